// RelationAwareStructureMatrix_18863496364257
// MI455X (gfx1250) — compile-verified
//
#include <hip/hip_runtime.h>
#include <hip/hip_bf16.h>

// Problem constants (match reference)
#define B_  2
#define H_  16
#define L_  1024
#define D_  64
#define R_  16
#define E_  4
#define BH_ (B_ * H_)

typedef __attribute__((ext_vector_type(2))) float v2f;
typedef __attribute__((ext_vector_type(8))) float v8f;

// ---------------------------------------------------------------------------
// Kernel 1: precompute low-rank projections
//   u[bh][l][r][e] = sum_d Q[bh][l][d] * A1[r][h][d][e]
//   v[bh][m][r][e] = sum_d K[bh][m][d] * A2[r][h][e][d]
// One 64-thread block per (mode, bh*L + row). Thread t -> (r = t/4, e = t%4).
// ---------------------------------------------------------------------------
__global__ __launch_bounds__(64) void uv_kernel(
    const float* __restrict__ Q, const float* __restrict__ K,
    const float* __restrict__ A1, const float* __restrict__ A2,
    float* __restrict__ u, float* __restrict__ v) {
  __shared__ __align__(16) float qs[D_];
  const int g    = blockIdx.x;      // bh * L + row
  const int mode = blockIdx.y;      // 0 -> u, 1 -> v
  const int tid  = threadIdx.x;
  const int bh   = g / L_;
  const int h    = bh % H_;
  const int r    = tid >> 2;
  const int e    = tid & 3;

  const float* src = (mode == 0) ? Q : K;
  qs[tid] = src[g * D_ + tid];
  __syncthreads();

  float acc = 0.f;
  if (mode == 0) {
    // A1[r][h][d][e], d fastest-but-one (stride E_)
    const float* a = A1 + ((r * H_ + h) * D_) * E_ + e;
#pragma unroll 8
    for (int d = 0; d < D_; ++d) acc += qs[d] * a[d * E_];
    u[g * (R_ * E_) + tid] = acc;
  } else {
    // A2[r][h][e][d], contiguous in d
    const float* a = A2 + ((r * H_ + h) * E_ + e) * D_;
#pragma unroll 8
    for (int d = 0; d < D_; ++d) acc += qs[d] * a[d];
    v[g * (R_ * E_) + tid] = acc;
  }
}

// ---------------------------------------------------------------------------
// Kernel 2: base = Q K^T via V_WMMA_F32_16X16X4_F32, plus gathered low-rank
// term and validity mask. 8 waves/block; wave w owns m-tile (blockIdx.x*8+w),
// all waves share l-tile blockIdx.y, bh = blockIdx.z.
//
// WMMA f32 16x16x4 lane layout (ISA 7.12.2):
//   A (16x4):  lane ln -> M = ln&15, VGPR j holds K = 2*(ln>>4) + j
//   B (4x16):  lane ln -> N = ln&15, VGPR j holds K = 2*(ln>>4) + j (mirrored)
//   C/D:       VGPR i, lane ln -> M = i + 8*(ln>>4), N = ln&15
// ---------------------------------------------------------------------------
__global__ __launch_bounds__(256) void score_kernel(
    const float* __restrict__ Q, const float* __restrict__ K,
    const int* __restrict__ rid, const float* __restrict__ u,
    const float* __restrict__ v, float* __restrict__ out) {
  __shared__ __align__(16) float u_s[16 * R_ * E_];        //  4 KB
  __shared__ __align__(16) float v_s[8 * 16 * R_ * E_];    // 32 KB

  const int tid = threadIdx.x;
  const int bh  = blockIdx.z;
  const int l0  = blockIdx.y * 16;
  const int mg  = blockIdx.x * 8;          // first m-tile handled by block
  const int b   = bh / H_;

  // Cooperative LDS fill: u slice (1024 floats) + 8 contiguous v tiles (8192).
  {
    const float4* usrc = (const float4*)(u + (bh * L_ + l0) * (R_ * E_));
    ((float4*)u_s)[tid] = usrc[tid];
    const float4* vsrc = (const float4*)(v + (bh * L_ + mg * 16) * (R_ * E_));
#pragma unroll
    for (int i = 0; i < 8; ++i)
      ((float4*)v_s)[tid + i * 256] = vsrc[tid + i * 256];
  }
  __syncthreads();

  const int wave = tid >> 5;
  const int ln   = tid & 31;
  const int row  = ln & 15;   // A-matrix M / B-matrix N for fragment loads
  const int hi   = ln >> 4;   // selects K sub-pair within each 4-wide chunk
  const int m0   = (mg + wave) * 16;

  const float* qrow = Q + (bh * L_ + l0 + row) * D_ + 2 * hi;
  const float* krow = K + (bh * L_ + m0 + row) * D_ + 2 * hi;

  v8f c = {};
#pragma unroll
  for (int kk = 0; kk < D_; kk += 4) {
    v2f a, bf;
    a[0]  = qrow[kk];
    a[1]  = qrow[kk + 1];
    bf[0] = krow[kk];
    bf[1] = krow[kk + 1];
    // D = A x B + C ; emits v_wmma_f32_16x16x4_f32
    c = __builtin_amdgcn_wmma_f32_16x16x4_f32(
        /*neg_a=*/false, a, /*neg_b=*/false, bf,
        /*c_mod=*/(short)0, c, /*reuse_a=*/false, /*reuse_b=*/false);
  }

  // Epilogue: gather per-(l,m) relation projections from LDS, mask, store.
  const int col = ln & 15;
  const float* vloc = v_s + wave * (16 * R_ * E_);
#pragma unroll
  for (int i = 0; i < 8; ++i) {
    const int r_in = i + 8 * hi;           // row within the 16x16 tile
    const int gl = l0 + r_in;
    const int gm = m0 + col;
    const int id = rid[(b * L_ + gl) * L_ + gm];
    float val = 0.f;
    if (id >= 0 && id < R_) {
      const float4 uu = *(const float4*)(u_s + r_in * (R_ * E_) + id * E_);
      const float4 vv = *(const float4*)(vloc + col * (R_ * E_) + id * E_);
      val = c[i] + uu.x * vv.x + uu.y * vv.y + uu.z * vv.z + uu.w * vv.w;
    }
    out[(bh * L_ + gl) * L_ + gm] = val;
  }
}

// ---------------------------------------------------------------------------
extern "C" void kernel_launch(void* const* d_in, const int* in_sizes, int n_in,
                              void* d_out, int out_size, void* d_ws, size_t ws_size,
                              hipStream_t stream) {
  const float* Q   = (const float*)d_in[0];
  const float* K   = (const float*)d_in[1];
  const int*   rid = (const int*)d_in[2];
  const float* A1  = (const float*)d_in[3];
  const float* A2  = (const float*)d_in[4];
  float* out = (float*)d_out;

  float* u = (float*)d_ws;                          // BH*L*R*E floats = 8 MB
  float* v = u + (size_t)BH_ * L_ * R_ * E_;        // another 8 MB

  // Stage 1: low-rank projections (u from Q/A1, v from K/A2).
  uv_kernel<<<dim3(BH_ * L_, 2), 64, 0, stream>>>(Q, K, A1, A2, u, v);

  // Stage 2: WMMA base GEMM + gathered low-rank epilogue.
  //   grid.x: groups of 8 m-tiles (64/8), grid.y: l-tiles (64), grid.z: B*H.
  score_kernel<<<dim3(L_ / 16 / 8, L_ / 16, BH_), 256, 0, stream>>>(
      Q, K, rid, u, v, out);
}